// UniPhyBlock_26010321945365
// MI455X (gfx1250) — compile-verified
//
#include <hip/hip_runtime.h>
#include <hip/hip_bf16.h>
#include <math.h>

// ---------------------------------------------------------------------------
// UniPhyBlock forward for MI455X (gfx1250), wave32 + WMMA.
// Pipeline: conv3x3 residual -> spectral mixer (DFT2 in LDS) ->
//           rfft2-domain SSM scan -> fused Port-Hamiltonian layer (WMMA).
// ---------------------------------------------------------------------------

constexpr int kB  = 2;
constexpr int kT  = 8;
constexpr int kC  = 16;
constexpr int kH  = 64;
constexpr int kW  = 64;
constexpr int kHW = kH * kW;                  // 4096
constexpr int kNE = kB * kT * kC * kHW;       // 1,048,576
constexpr int kD  = 2 * kC;                   // 32
constexpr int kHid = kD * 4;                  // 128
constexpr int kNPOS = kB * kT * kHW;          // 65,536 spatial positions

typedef float v2f __attribute__((ext_vector_type(2)));
typedef float v8f __attribute__((ext_vector_type(8)));

// fp32 WMMA 16x16x4: D = A(16x4) * B(4x16) + C  (wave32, per-lane v2f/v2f/v8f)
static __device__ __forceinline__ v8f wmma4(v2f a, v2f b, v8f c) {
  return __builtin_amdgcn_wmma_f32_16x16x4_f32(
      /*neg_a=*/false, a, /*neg_b=*/false, b,
      /*c_mod=*/(short)0, c, /*reuse_a=*/false, /*reuse_b=*/false);
}

// ---------------------------------------------------------------------------
// Prep: dt_eff, dt_scalar, A = exp(lam * dt_eff), noise scale.
// prep layout: [0..15]=A_re, [16..31]=A_im, [32]=dt_scalar, [33]=sigma*sqrt(dt)
// ---------------------------------------------------------------------------
__global__ void prep_kernel(const float* dt, const float* lam_re,
                            const float* lam_im, const float* dt_param,
                            const float* sigma, float* prep) {
  if (threadIdx.x == 0 && blockIdx.x == 0) {
    float dte[kC];
    float dts = 0.f;
    for (int c = 0; c < kC; ++c) {
      float p = dt_param[c];
      float sp = (p > 20.f) ? p : log1pf(expf(p));   // softplus
      dte[c] = dt[0] * sp;
      dts += dte[c];
    }
    dts *= (1.f / kC);
    for (int c = 0; c < kC; ++c) {
      float lr = lam_re[c];
      float sp = (lr > 20.f) ? lr : log1pf(expf(lr));
      float re = -sp * dte[c];
      float im = lam_im[c] * dte[c];
      float m = expf(re);
      prep[c]       = m * cosf(im);
      prep[kC + c]  = m * sinf(im);
    }
    prep[32] = dts;
    prep[33] = sigma[0] * sqrtf(dts);
  }
}

// ---------------------------------------------------------------------------
// Conv: x1 = x + (conv3x3(x) + bias), applied to real & imag fields.
// ---------------------------------------------------------------------------
__global__ void conv_kernel(const float* __restrict__ xr,
                            const float* __restrict__ xi,
                            const float* __restrict__ w,
                            const float* __restrict__ bias,
                            float* __restrict__ cr, float* __restrict__ cimag) {
  int tid = blockIdx.x * blockDim.x + threadIdx.x;   // [0, 2*kNE)
  int f = tid / kNE;
  int r = tid - f * kNE;
  int bt  = r / (kC * kHW);
  int r2  = r - bt * (kC * kHW);
  int co  = r2 / kHW;
  int pos = r2 - co * kHW;
  int h = pos >> 6, wc = pos & 63;
  const float* field = (f == 0) ? xr : xi;
  const float* in = field + (size_t)bt * kC * kHW;
  float acc = bias[co];
  for (int cc = 0; cc < kC; ++cc) {
    const float* ip = in + cc * kHW;
    const float* wp = w + (co * kC + cc) * 9;
    for (int kh = 0; kh < 3; ++kh) {
      int hh = h + kh - 1;
      if (hh < 0 || hh >= kH) continue;
      for (int kw = 0; kw < 3; ++kw) {
        int ww = wc + kw - 1;
        if (ww < 0 || ww >= kW) continue;
        acc += wp[kh * 3 + kw] * ip[hh * kW + ww];
      }
    }
  }
  float* out = (f == 0) ? cr : cimag;
  out[r] = field[r] + acc;
}

// ---------------------------------------------------------------------------
// In-LDS 64x64 complex DFT2 (ortho). sgn=-1 forward, +1 inverse.
// 256 threads cooperate; a/b are 4096-float LDS arrays; twc/tws 64-entry table.
// ---------------------------------------------------------------------------
static __device__ void dft2_inplace(float* ar, float* ai, float* br, float* bi,
                                    const float* twc, const float* tws,
                                    float sgn) {
  const int t = threadIdx.x;
  // pass 1: rows, a -> b   (out[r,k] = sum_n a[r,n] * e^{sgn*i*2pi*nk/64} / 8)
  for (int o = 0; o < 16; ++o) {
    int idx = t + o * 256;
    int r = idx >> 6, k = idx & 63;
    const float* pr = ar + (r << 6);
    const float* pi = ai + (r << 6);
    float sre = 0.f, sim = 0.f;
    for (int n = 0; n < 64; ++n) {
      int tw = (n * k) & 63;
      float c = twc[tw], s = sgn * tws[tw];
      float vr = pr[n], vi = pi[n];
      sre += vr * c - vi * s;
      sim += vr * s + vi * c;
    }
    br[idx] = 0.125f * sre;
    bi[idx] = 0.125f * sim;
  }
  __syncthreads();
  // pass 2: columns, b -> a (out[k,col] = sum_n b[n,col] * tw(nk) / 8)
  for (int o = 0; o < 16; ++o) {
    int idx = t + o * 256;
    int k = idx >> 6, col = idx & 63;
    float sre = 0.f, sim = 0.f;
    for (int n = 0; n < 64; ++n) {
      int tw = (n * k) & 63;
      float c = twc[tw], s = sgn * tws[tw];
      float vr = br[(n << 6) + col], vi = bi[(n << 6) + col];
      sre += vr * c - vi * s;
      sim += vr * s + vi * c;
    }
    ar[idx] = 0.125f * sre;
    ai[idx] = 0.125f * sim;
  }
  __syncthreads();
}

static __device__ __forceinline__ void init_twiddle(float* twc, float* tws) {
  int t = threadIdx.x;
  if (t < 64) {
    float ang = 6.283185307179586f * (float)t / 64.f;
    twc[t] = cosf(ang);
    tws[t] = sinf(ang);
  }
}

// ---------------------------------------------------------------------------
// Spectral mixer: x2 = x1 + ifft2(fft2(x1) * filt).  One WG per (bt,c) image.
// ---------------------------------------------------------------------------
__global__ void spec_kernel(const float* __restrict__ cr,
                            const float* __restrict__ cimag,
                            const float* __restrict__ fre,
                            const float* __restrict__ fim,
                            float* __restrict__ sr, float* __restrict__ si) {
  extern __shared__ float sm[];
  float* ar = sm;           float* ai = sm + 4096;
  float* br = sm + 8192;    float* bi = sm + 12288;
  float* twc = sm + 16384;  float* tws = sm + 16448;
  init_twiddle(twc, tws);
  int t = threadIdx.x;
  int img = blockIdx.x;                   // (b*T+t)*C + c
  int c = img % kC;
  size_t base = (size_t)img * kHW;
  for (int o = 0; o < 16; ++o) {
    int i = t + o * 256;
    ar[i] = cr[base + i];
    ai[i] = cimag[base + i];
  }
  __syncthreads();
  dft2_inplace(ar, ai, br, bi, twc, tws, -1.f);
  const float* fr = fre + (size_t)c * kHW;
  const float* fi = fim + (size_t)c * kHW;
  for (int o = 0; o < 16; ++o) {
    int i = t + o * 256;
    float vr = ar[i], vi = ai[i], r = fr[i], m = fi[i];
    ar[i] = vr * r - vi * m;
    ai[i] = vr * m + vi * r;
  }
  __syncthreads();
  dft2_inplace(ar, ai, br, bi, twc, tws, 1.f);
  for (int o = 0; o < 16; ++o) {
    int i = t + o * 256;
    sr[base + i] = cr[base + i] + ar[i];
    si[base + i] = cimag[base + i] + ai[i];
  }
}

// ---------------------------------------------------------------------------
// Forward fft2 of real field (f=0: x2.real, f=1: x2.imag) -> full spectrum.
// ---------------------------------------------------------------------------
__global__ void fftR_kernel(const float* __restrict__ sr,
                            const float* __restrict__ si,
                            float* __restrict__ Sr, float* __restrict__ Si) {
  extern __shared__ float sm[];
  float* ar = sm;           float* ai = sm + 4096;
  float* br = sm + 8192;    float* bi = sm + 12288;
  float* twc = sm + 16384;  float* tws = sm + 16448;
  init_twiddle(twc, tws);
  int t = threadIdx.x;
  int g = blockIdx.x;                 // [0,512): f*256 + img
  int f = g >> 8;
  int img = g & 255;
  const float* src = (f == 0) ? sr : si;
  size_t base = (size_t)img * kHW;
  for (int o = 0; o < 16; ++o) {
    int i = t + o * 256;
    ar[i] = src[base + i];
    ai[i] = 0.f;
  }
  __syncthreads();
  dft2_inplace(ar, ai, br, bi, twc, tws, -1.f);
  size_t ob = (size_t)f * kNE + base;
  for (int o = 0; o < 16; ++o) {
    int i = t + o * 256;
    Sr[ob + i] = ar[i];
    Si[ob + i] = ai[i];
  }
}

// ---------------------------------------------------------------------------
// SSM scan over T (in-place, per frequency bin). Pole A for kx<=32 (rfft half),
// conj(A) for the hermitian mirror kx>32 -> Re(ifft2) == irfft2(half scan).
// ---------------------------------------------------------------------------
__global__ void scan_kernel(float* __restrict__ Sr, float* __restrict__ Si,
                            const float* __restrict__ prep) {
  int tid = blockIdx.x * blockDim.x + threadIdx.x;  // [0, 2*kB*kC*kHW)
  int f = tid / (kB * kC * kHW);
  int r = tid - f * (kB * kC * kHW);
  int b = r / (kC * kHW);
  int r2 = r - b * (kC * kHW);
  int c = r2 / kHW;
  int pos = r2 - c * kHW;
  int kx = pos & 63;
  float are = prep[c];
  float aim = prep[kC + c];
  if (kx > 32) aim = -aim;
  size_t idx = (size_t)f * kNE + ((size_t)(b * kT) * kC + c) * kHW + pos;
  size_t stride = (size_t)kC * kHW;
  float hr = 0.f, hi = 0.f;
  for (int tt = 0; tt < kT; ++tt) {
    float ur = Sr[idx], ui = Si[idx];
    float nr = are * hr - aim * hi + ur;
    float ni = are * hi + aim * hr + ui;
    hr = nr; hi = ni;
    Sr[idx] = hr;
    Si[idx] = hi;
    idx += stride;
  }
}

// ---------------------------------------------------------------------------
// Inverse fft2, take real part, combine: x3 = x2 + Re(ifft2(h)) + ns*noise.
// ---------------------------------------------------------------------------
__global__ void ifftC_kernel(const float* __restrict__ Sr,
                             const float* __restrict__ Si,
                             const float* __restrict__ sr,
                             const float* __restrict__ si,
                             const float* __restrict__ nzr,
                             const float* __restrict__ nzi,
                             const float* __restrict__ prep,
                             float* __restrict__ x3r, float* __restrict__ x3i) {
  extern __shared__ float sm[];
  float* ar = sm;           float* ai = sm + 4096;
  float* br = sm + 8192;    float* bi = sm + 12288;
  float* twc = sm + 16384;  float* tws = sm + 16448;
  init_twiddle(twc, tws);
  int t = threadIdx.x;
  int g = blockIdx.x;
  int f = g >> 8;
  int img = g & 255;
  size_t sbase = (size_t)f * kNE + (size_t)img * kHW;
  for (int o = 0; o < 16; ++o) {
    int i = t + o * 256;
    ar[i] = Sr[sbase + i];
    ai[i] = Si[sbase + i];
  }
  __syncthreads();
  dft2_inplace(ar, ai, br, bi, twc, tws, 1.f);
  float ns = prep[33];
  const float* xin = (f == 0) ? sr : si;
  const float* nz  = (f == 0) ? nzr : nzi;
  float* out = (f == 0) ? x3r : x3i;
  size_t base = (size_t)img * kHW;
  for (int o = 0; o < 16; ++o) {
    int i = t + o * 256;
    out[base + i] = xin[base + i] + ar[i] + ns * nz[base + i];
  }
}

// ---------------------------------------------------------------------------
// Fused Port-Hamiltonian layer (the WMMA workhorse).
// Block = 128 threads (4 waves); each wave owns 16 consecutive positions.
// update = M g - M^T g - P^T(P g) - 1e-4 g.  M and P never touch LDS/HBM:
// each WMMA D-tile (fixed matrix-row i, 16-wide j block, 16 positions) is
// consumed immediately:
//   (M g)[p][i]   : per-lane D*g[p][j] + 4-step shfl_xor reduction (width 16)
//   (M^T g)[p][j] : per-lane register FMA vs broadcast g[p][i]
//   P g           : same as (M g);  P^T(Pg): recompute P pass vs broadcast Pg
// Per-wave LDS ~18.6 KB -> 4 workgroups (16 waves) per WGP.
// ---------------------------------------------------------------------------
__global__ void __launch_bounds__(128) ph_kernel(
    const float* __restrict__ x3r, const float* __restrict__ x3i,
    const float* __restrict__ ln_g, const float* __restrict__ ln_b,
    const float* __restrict__ W1, const float* __restrict__ b1,
    const float* __restrict__ w2, const float* __restrict__ JgW,
    const float* __restrict__ Jgb, const float* __restrict__ RgW,
    const float* __restrict__ Rgb, const float* __restrict__ prep,
    float* __restrict__ out) {
  __shared__ float Xin[4][16][33];    // layernormed inputs (pad: stride 33)
  __shared__ float Uact[4][16][132];  // silu'(z)*w2       (pad: stride 132)
  __shared__ float Gv[4][16][33];     // potential gradient
  __shared__ float xcs[4][16][33];    // raw xc for residual
  __shared__ float MgL[4][16][33];    // Mg accum -> final update
  __shared__ float PgL[4][16][33];    // Pg accum

  const int wv = threadIdx.x >> 5;
  const int lane = threadIdx.x & 31;
  const int lo = lane & 15;
  const int hi = lane >> 4;
  const int tile = blockIdx.x * 4 + wv;
  const int p0 = tile * 16;           // 16-aligned; never straddles bt
  const int bt = p0 >> 12;
  const int hw0 = p0 & 4095;
  const float dts = prep[32];

  __builtin_prefetch(JgW + (threadIdx.x << 4), 0, 0);
  __builtin_prefetch(RgW + (threadIdx.x << 4), 0, 0);

  // ---- coalesced gather: lane lo = position, hi selects real/imag half ----
  {
    const float* src = hi ? x3i : x3r;
    for (int dh = 0; dh < 16; ++dh) {
      xcs[wv][lo][hi * 16 + dh] =
          src[((size_t)bt * kC + dh) * kHW + hw0 + lo];
    }
  }
  __syncthreads();

  // ---- layernorm (per position p=lo; stats computed redundantly per hi) ----
  {
    float mu = 0.f;
    for (int d = 0; d < kD; ++d) mu += xcs[wv][lo][d];
    mu *= (1.f / kD);
    float var = 0.f;
    for (int d = 0; d < kD; ++d) {
      float dd = xcs[wv][lo][d] - mu;
      var += dd * dd;
    }
    var *= (1.f / kD);
    float inv = rsqrtf(var + 1e-5f);
    for (int r = 0; r < 16; ++r) {
      int d = hi * 16 + r;
      Xin[wv][lo][d] = (xcs[wv][lo][d] - mu) * inv * ln_g[d] + ln_b[d];
    }
  }
  __syncthreads();

  // ---- Z = X @ W1 + b1 ; Uact = silu'(Z) * w2 ----
  for (int nt = 0; nt < 8; ++nt) {
    int col = nt * 16 + lo;
    v8f acc;
    float bb = b1[col];
#pragma unroll
    for (int vv = 0; vv < 8; ++vv) acc[vv] = bb;
#pragma unroll
    for (int kk = 0; kk < 8; ++kk) {
      int k = kk * 4 + 2 * hi;
      v2f a, b;
      a.x = Xin[wv][lo][k];      a.y = Xin[wv][lo][k + 1];
      b.x = W1[k * kHid + col];  b.y = W1[(k + 1) * kHid + col];
      acc = wmma4(a, b, acc);
    }
    float wcol = w2[col];
#pragma unroll
    for (int vv = 0; vv < 8; ++vv) {
      float z = acc[vv];
      float sg = 1.f / (1.f + expf(-z));
      float dsilu = sg * (1.f + z * (1.f - sg));
      Uact[wv][vv + 8 * hi][col] = dsilu * wcol;
    }
  }
  __syncthreads();

  // ---- G = Uact @ W1^T ----
  for (int nt = 0; nt < 2; ++nt) {
    int col = nt * 16 + lo;
    v8f acc;
#pragma unroll
    for (int vv = 0; vv < 8; ++vv) acc[vv] = 0.f;
#pragma unroll
    for (int kk = 0; kk < 32; ++kk) {
      int k = kk * 4 + 2 * hi;
      v2f a, b;
      a.x = Uact[wv][lo][k];     a.y = Uact[wv][lo][k + 1];
      b.x = W1[col * kHid + k];  b.y = W1[col * kHid + k + 1];
      acc = wmma4(a, b, acc);
    }
#pragma unroll
    for (int vv = 0; vv < 8; ++vv) Gv[wv][vv + 8 * hi][col] = acc[vv];
  }
  __syncthreads();

  // ---- preload g registers; zero accumulators ----
  float greg[2][8];   // g[p = vv+8*hi][j = par*16+lo]
  float mtg[2][8];    // (M^T g)[p][j], same ownership
#pragma unroll
  for (int par = 0; par < 2; ++par)
#pragma unroll
    for (int vv = 0; vv < 8; ++vv) {
      greg[par][vv] = Gv[wv][vv + 8 * hi][par * 16 + lo];
      mtg[par][vv] = 0.f;
      MgL[wv][vv + 8 * hi][par * 16 + lo] = 0.f;
      PgL[wv][vv + 8 * hi][par * 16 + lo] = 0.f;
    }
  __syncthreads();

  // ---- M pass: M = X@Jg_W + Jg_b, fused Mg (shfl-reduce) and M^T g ----
  for (int i2 = 0; i2 < 32; ++i2) {
    float gbc[8];
#pragma unroll
    for (int vv = 0; vv < 8; ++vv) gbc[vv] = Gv[wv][vv + 8 * hi][i2];
#pragma unroll
    for (int par = 0; par < 2; ++par) {
      int col = i2 * 32 + par * 16 + lo;
      v8f acc;
      float bb = Jgb[col];
#pragma unroll
      for (int vv = 0; vv < 8; ++vv) acc[vv] = bb;
#pragma unroll
      for (int kk = 0; kk < 8; ++kk) {
        int k = kk * 4 + 2 * hi;
        v2f a, b;
        a.x = Xin[wv][lo][k];       a.y = Xin[wv][lo][k + 1];
        b.x = JgW[k * 1024 + col];  b.y = JgW[(k + 1) * 1024 + col];
        acc = wmma4(a, b, acc);
      }
#pragma unroll
      for (int vv = 0; vv < 8; ++vv) {
        float part = acc[vv] * greg[par][vv];
        part += __shfl_xor(part, 1, 16);
        part += __shfl_xor(part, 2, 16);
        part += __shfl_xor(part, 4, 16);
        part += __shfl_xor(part, 8, 16);
        if (lo == vv) MgL[wv][vv + 8 * hi][i2] += part;
        mtg[par][vv] += acc[vv] * gbc[vv];
      }
    }
  }
  __syncthreads();

  // ---- P pass 1: P = X@Rg_W + Rg_b, fused Pg (shfl-reduce) ----
  for (int i2 = 0; i2 < 32; ++i2) {
#pragma unroll
    for (int par = 0; par < 2; ++par) {
      int col = i2 * 32 + par * 16 + lo;
      v8f acc;
      float bb = Rgb[col];
#pragma unroll
      for (int vv = 0; vv < 8; ++vv) acc[vv] = bb;
#pragma unroll
      for (int kk = 0; kk < 8; ++kk) {
        int k = kk * 4 + 2 * hi;
        v2f a, b;
        a.x = Xin[wv][lo][k];       a.y = Xin[wv][lo][k + 1];
        b.x = RgW[k * 1024 + col];  b.y = RgW[(k + 1) * 1024 + col];
        acc = wmma4(a, b, acc);
      }
#pragma unroll
      for (int vv = 0; vv < 8; ++vv) {
        float part = acc[vv] * greg[par][vv];
        part += __shfl_xor(part, 1, 16);
        part += __shfl_xor(part, 2, 16);
        part += __shfl_xor(part, 4, 16);
        part += __shfl_xor(part, 8, 16);
        if (lo == vv) PgL[wv][vv + 8 * hi][i2] += part;
      }
    }
  }
  __syncthreads();

  // ---- P pass 2: recompute P, accumulate P^T (Pg) vs broadcast Pg ----
  float ptpg[2][8];
#pragma unroll
  for (int par = 0; par < 2; ++par)
#pragma unroll
    for (int vv = 0; vv < 8; ++vv) ptpg[par][vv] = 0.f;
  for (int i2 = 0; i2 < 32; ++i2) {
    float pbc[8];
#pragma unroll
    for (int vv = 0; vv < 8; ++vv) pbc[vv] = PgL[wv][vv + 8 * hi][i2];
#pragma unroll
    for (int par = 0; par < 2; ++par) {
      int col = i2 * 32 + par * 16 + lo;
      v8f acc;
      float bb = Rgb[col];
#pragma unroll
      for (int vv = 0; vv < 8; ++vv) acc[vv] = bb;
#pragma unroll
      for (int kk = 0; kk < 8; ++kk) {
        int k = kk * 4 + 2 * hi;
        v2f a, b;
        a.x = Xin[wv][lo][k];       a.y = Xin[wv][lo][k + 1];
        b.x = RgW[k * 1024 + col];  b.y = RgW[(k + 1) * 1024 + col];
        acc = wmma4(a, b, acc);
      }
#pragma unroll
      for (int vv = 0; vv < 8; ++vv) ptpg[par][vv] += acc[vv] * pbc[vv];
    }
  }
  __syncthreads();

  // ---- finalize update in MgL: upd = Mg - M^T g - P^T Pg - 1e-4 g ----
#pragma unroll
  for (int par = 0; par < 2; ++par)
#pragma unroll
    for (int vv = 0; vv < 8; ++vv) {
      int p = vv + 8 * hi;
      int j = par * 16 + lo;
      MgL[wv][p][j] = MgL[wv][p][j] - mtg[par][vv] - ptpg[par][vv] -
                      1e-4f * greg[par][vv];
    }
  __syncthreads();

  // ---- coalesced output: lane lo = position, hi splits channel range ----
  for (int r = 0; r < 16; ++r) {
    int d = hi * 16 + r;
    out[((size_t)bt * kD + d) * kHW + hw0 + lo] =
        xcs[wv][lo][d] + dts * MgL[wv][lo][d];
  }
}

// ---------------------------------------------------------------------------
// Launch
// ---------------------------------------------------------------------------
extern "C" void kernel_launch(void* const* d_in, const int* in_sizes, int n_in,
                              void* d_out, int out_size, void* d_ws,
                              size_t ws_size, hipStream_t stream) {
  (void)in_sizes; (void)n_in; (void)out_size; (void)ws_size;
  const float* x_real   = (const float*)d_in[0];
  const float* x_imag   = (const float*)d_in[1];
  const float* dt       = (const float*)d_in[2];
  const float* conv_w   = (const float*)d_in[3];
  const float* conv_b   = (const float*)d_in[4];
  const float* spec_re  = (const float*)d_in[5];
  const float* spec_im  = (const float*)d_in[6];
  const float* lam_re   = (const float*)d_in[7];
  const float* lam_im   = (const float*)d_in[8];
  const float* dt_param = (const float*)d_in[9];
  const float* sigma    = (const float*)d_in[10];
  const float* noise_r  = (const float*)d_in[11];
  const float* noise_i  = (const float*)d_in[12];
  const float* ln_g     = (const float*)d_in[13];
  const float* ln_b     = (const float*)d_in[14];
  const float* W1       = (const float*)d_in[15];
  const float* b1       = (const float*)d_in[16];
  const float* w2       = (const float*)d_in[17];
  const float* JgW      = (const float*)d_in[19];
  const float* Jgb      = (const float*)d_in[20];
  const float* RgW      = (const float*)d_in[21];
  const float* Rgb      = (const float*)d_in[22];

  float* ws  = (float*)d_ws;
  float* cr   = ws;                     // x1.real
  float* cim  = ws + (size_t)kNE;       // x1.imag
  float* sr   = ws + (size_t)2 * kNE;   // x2.real
  float* si   = ws + (size_t)3 * kNE;   // x2.imag
  float* x3r  = ws + (size_t)4 * kNE;   // pre-PH real
  float* x3i  = ws + (size_t)5 * kNE;   // pre-PH imag
  float* Sr   = ws + (size_t)6 * kNE;   // spectra real (2 fields)
  float* Si   = ws + (size_t)8 * kNE;   // spectra imag (2 fields)
  float* prep = ws + (size_t)10 * kNE;  // 64 floats

  const size_t fft_smem = (size_t)(16384 + 128) * sizeof(float);  // 66,048 B

  prep_kernel<<<1, 64, 0, stream>>>(dt, lam_re, lam_im, dt_param, sigma, prep);
  conv_kernel<<<(2 * kNE) / 256, 256, 0, stream>>>(x_real, x_imag, conv_w,
                                                   conv_b, cr, cim);
  spec_kernel<<<kB * kT * kC, 256, fft_smem, stream>>>(cr, cim, spec_re,
                                                       spec_im, sr, si);
  fftR_kernel<<<2 * kB * kT * kC, 256, fft_smem, stream>>>(sr, si, Sr, Si);
  scan_kernel<<<(2 * kB * kC * kHW) / 256, 256, 0, stream>>>(Sr, Si, prep);
  ifftC_kernel<<<2 * kB * kT * kC, 256, fft_smem, stream>>>(
      Sr, Si, sr, si, noise_r, noise_i, prep, x3r, x3i);
  ph_kernel<<<kNPOS / 64, 128, 0, stream>>>(
      x3r, x3i, ln_g, ln_b, W1, b1, w2, JgW, Jgb, RgW, Rgb, prep,
      (float*)d_out);
}